// TripletLoss_6605659701626
// MI455X (gfx1250) — compile-verified
//
#include <hip/hip_runtime.h>
#include <hip/hip_bf16.h>
#include <math.h>

#define B_SZ   1024
#define D_SZ   2048
#define MARGIN 0.3f
#define EPS_F  1e-12f
#define NGRP   8              // column groups (blockIdx.y)

typedef float v2f __attribute__((ext_vector_type(2)));
typedef float v8f __attribute__((ext_vector_type(8)));

// ---------------------------------------------------------------------------
// Kernel 1: row squared norms. 8 waves/block, one row per wave.
// ---------------------------------------------------------------------------
__global__ __launch_bounds__(256) void norms_kernel(const float* __restrict__ F,
                                                    float* __restrict__ sq) {
    const int wave = threadIdx.x >> 5;
    const int lane = threadIdx.x & 31;
    const int row  = blockIdx.x * 8 + wave;
    const float4* Frow = (const float4*)(F + (size_t)row * D_SZ);
    float s = 0.0f;
#pragma unroll
    for (int i = 0; i < D_SZ / 128; ++i) {       // 16 iters * 32 lanes * 4 = 2048
        float4 v = Frow[i * 32 + lane];
        s += v.x * v.x + v.y * v.y + v.z * v.z + v.w * v.w;
    }
#pragma unroll
    for (int m = 16; m >= 1; m >>= 1) s += __shfl_xor(s, m, 32);
    if (lane == 0) sq[row] = s;
}

// ---------------------------------------------------------------------------
// Kernel 2: fused Gram tile (f32 WMMA) + masked hardest-pos/neg row reduce.
// grid = (64 row-strips, NGRP col groups); 8 waves/block, 1 16x16 tile/wave.
// Inner K loop is double-buffered: 8 (a,b) pairs (32 K) load while the other
// buffer feeds 8 back-to-back WMMAs, hiding global-load latency.
// ---------------------------------------------------------------------------
__global__ __launch_bounds__(256) void tile_kernel(const float* __restrict__ F,
                                                   const int*   __restrict__ labels,
                                                   const float* __restrict__ sq,
                                                   float* __restrict__ hp_part,   // [NGRP][B_SZ]
                                                   float* __restrict__ hn_part) { // [NGRP][B_SZ]
    __shared__ float lds_hp[8][16];
    __shared__ float lds_hn[8][16];

    const int wave = threadIdx.x >> 5;
    const int lane = threadIdx.x & 31;
    const int lo   = lane & 15;      // N (and M for loads)
    const int hi   = lane >> 4;      // selects K pair / row half
    const int rowBase = blockIdx.x * 16;
    const int colBase = (blockIdx.y * 8 + wave) * 16;

    // Per-lane metadata for rows M = 8*hi + r (uniform across each 16-lane half)
    float sqR[8]; int labR[8];
#pragma unroll
    for (int r = 0; r < 8; ++r) {
        const int gr = rowBase + 8 * hi + r;
        sqR[r]  = sq[gr];
        labR[r] = labels[gr];
    }

    const float* aptr = F + (size_t)(rowBase + lo) * D_SZ + 2 * hi;
    const float* bptr = F + (size_t)(colBase + lo) * D_SZ + 2 * hi;

    v8f acc = {0.f, 0.f, 0.f, 0.f, 0.f, 0.f, 0.f, 0.f};

    // --- software-pipelined K loop, 64 K per outer iteration -------------
    v2f A0[8], B0[8], A1[8], B1[8];
#pragma unroll
    for (int i = 0; i < 8; ++i) {                 // prologue: K[0..32)
        A0[i] = *(const v2f*)(aptr + 4 * i);
        B0[i] = *(const v2f*)(bptr + 4 * i);
    }

    for (int k = 0; k < D_SZ; k += 64) {
        // prefetch K[k+32 .. k+64) while buf0 computes
#pragma unroll
        for (int i = 0; i < 8; ++i) {
            A1[i] = *(const v2f*)(aptr + k + 32 + 4 * i);
            B1[i] = *(const v2f*)(bptr + k + 32 + 4 * i);
        }
#pragma unroll
        for (int i = 0; i < 8; ++i)
            acc = __builtin_amdgcn_wmma_f32_16x16x4_f32(false, A0[i], false, B0[i],
                                                        (short)0, acc, false, false);
        // prefetch K[k+64 .. k+96) while buf1 computes (scalar-uniform guard)
        if (k + 64 < D_SZ) {
#pragma unroll
            for (int i = 0; i < 8; ++i) {
                A0[i] = *(const v2f*)(aptr + k + 64 + 4 * i);
                B0[i] = *(const v2f*)(bptr + k + 64 + 4 * i);
            }
        }
#pragma unroll
        for (int i = 0; i < 8; ++i)
            acc = __builtin_amdgcn_wmma_f32_16x16x4_f32(false, A1[i], false, B1[i],
                                                        (short)0, acc, false, false);
    }

    // Epilogue: distance + masking, per-element
    const float sqc  = sq[colBase + lo];
    const int   labC = labels[colBase + lo];
    const int   gcol = colBase + lo;

    float hp[8], hn[8];
#pragma unroll
    for (int r = 0; r < 8; ++r) {
        const int grow = rowBase + 8 * hi + r;
        const float d2 = sqR[r] + sqc - 2.0f * acc[r];
        const float d  = sqrtf(fmaxf(d2, EPS_F));
        const bool same = (labR[r] == labC);
        const bool diag = (grow == gcol);
        hp[r] = (same && !diag) ? d : -INFINITY;
        hn[r] = (!same)         ? d :  INFINITY;
    }

    // Reduce across the 16 lanes of each half (masks <16 keep halves separate)
#pragma unroll
    for (int r = 0; r < 8; ++r) {
#pragma unroll
        for (int m = 8; m >= 1; m >>= 1) {
            hp[r] = fmaxf(hp[r], __shfl_xor(hp[r], m, 32));
            hn[r] = fminf(hn[r], __shfl_xor(hn[r], m, 32));
        }
    }

    if (lo == 0) {
#pragma unroll
        for (int r = 0; r < 8; ++r) {
            lds_hp[wave][8 * hi + r] = hp[r];
            lds_hn[wave][8 * hi + r] = hn[r];
        }
    }
    __syncthreads();

    // Combine the 8 waves of this block; write per-(strip, colgroup) partials
    if (threadIdx.x < 16) {
        const int m = threadIdx.x;
        float HP = -INFINITY, HN = INFINITY;
#pragma unroll
        for (int w = 0; w < 8; ++w) {
            HP = fmaxf(HP, lds_hp[w][m]);
            HN = fminf(HN, lds_hn[w][m]);
        }
        hp_part[blockIdx.y * B_SZ + rowBase + m] = HP;
        hn_part[blockIdx.y * B_SZ + rowBase + m] = HN;
    }
}

// ---------------------------------------------------------------------------
// Kernel 3: merge column-group partials, per-row loss, deterministic reduce.
// ---------------------------------------------------------------------------
__global__ __launch_bounds__(1024) void finalize_kernel(const float* __restrict__ hp_part,
                                                        const float* __restrict__ hn_part,
                                                        float* __restrict__ out) {
    __shared__ float s_per[1024];
    __shared__ int   s_val[1024];
    __shared__ int   s_act[1024];
    const int t = threadIdx.x;

    float HP = -INFINITY, HN = INFINITY;
#pragma unroll
    for (int g = 0; g < NGRP; ++g) {
        HP = fmaxf(HP, hp_part[g * B_SZ + t]);
        HN = fminf(HN, hn_part[g * B_SZ + t]);
    }
    const bool valid = (HP > -INFINITY) && (HN < INFINITY);
    const float per  = valid ? fmaxf(HP - HN + MARGIN, 0.0f) : 0.0f;

    s_per[t] = per;
    s_val[t] = valid ? 1 : 0;
    s_act[t] = (valid && per > 0.0f) ? 1 : 0;
    __syncthreads();

    for (int off = 512; off >= 1; off >>= 1) {
        if (t < off) {
            s_per[t] += s_per[t + off];
            s_val[t] += s_val[t + off];
            s_act[t] += s_act[t + off];
        }
        __syncthreads();
    }
    if (t == 0) {
        const int nv = (s_val[0] < 1) ? 1 : s_val[0];
        out[0] = s_per[0] / (float)nv;
        out[1] = (float)s_act[0];
    }
}

// ---------------------------------------------------------------------------
extern "C" void kernel_launch(void* const* d_in, const int* in_sizes, int n_in,
                              void* d_out, int out_size, void* d_ws, size_t ws_size,
                              hipStream_t stream) {
    const float* F      = (const float*)d_in[0];
    const int*   labels = (const int*)d_in[1];

    char* ws = (char*)d_ws;
    float* sq      = (float*)(ws);                              //  4 KB
    float* hp_part = (float*)(ws + 4 * 1024);                   // 32 KB
    float* hn_part = (float*)(ws + 4 * 1024 + NGRP * B_SZ * 4); // 32 KB

    norms_kernel<<<B_SZ / 8, 256, 0, stream>>>(F, sq);
    tile_kernel<<<dim3(B_SZ / 16, NGRP), 256, 0, stream>>>(F, labels, sq,
                                                           hp_part, hn_part);
    finalize_kernel<<<1, 1024, 0, stream>>>(hp_part, hn_part, (float*)d_out);
}